// SubsetSampling_33844342292791
// MI455X (gfx1250) — compile-verified
//
#include <hip/hip_runtime.h>
#include <cstdint>
#include <cstddef>

// ---------------------------------------------------------------------------
// Subset sampling (Gumbel top-k relaxation), B=64 rows, N=262144, K=16.
// Memory-bound streaming workload: no GEMM structure -> WMMA not applicable.
// CDNA5 paths used:
//   * global_load_async_to_lds_b128 (ASYNCcnt) staging, depth-3 pipeline
//   * tensor_load_to_lds (TDM, TENSORcnt) DMA staging in the first pass
//   * s_wait_asynccnt / s_wait_tensorcnt split counters, DS reductions
//   * MI455X 192MB L2 residency: NT stores for write-once outputs,
//     TH_LOAD_LU for last-use input reads (keeps logits+g+keys = 192MB hot)
// ---------------------------------------------------------------------------

namespace {
constexpr int   Brows = 64;
constexpr int   Ncols = 262144;
constexpr int   Ksel  = 16;
constexpr int   CH    = 32;            // chunks per row
constexpr int   CHUNK = Ncols / CH;    // 8192 elements per chunk
constexpr int   TPB   = 256;           // 8 wave32 per workgroup
constexpr int   TILE  = 1024;          // staged elements per tile (4 per thread)
constexpr int   NT    = CHUNK / TILE;  // 8 tiles per chunk
constexpr int   DEPTH = 3;             // async tiles in flight
constexpr int   NBUF  = 4;             // LDS buffers (DEPTH+1)
constexpr float EPS   = 1.17549435e-38f; // np.finfo(float32).tiny
}

typedef float f32x4 __attribute__((ext_vector_type(4)));

// ---- CDNA5 async global->LDS staging (ASYNCcnt) ----------------------------
__device__ __forceinline__ void async_b128(const float* gsrc, const float* ldst) {
  unsigned           lds  = (unsigned)(uintptr_t)ldst;       // low 32 bits = LDS offset
  unsigned long long gptr = (unsigned long long)(uintptr_t)gsrc;
  asm volatile("global_load_async_to_lds_b128 %0, %1, off"
               :: "v"(lds), "v"(gptr) : "memory");
}
// Last-use variant: NT + discard-on-hit, for final reads of streaming inputs.
__device__ __forceinline__ void async_b128_lu(const float* gsrc, const float* ldst) {
  unsigned           lds  = (unsigned)(uintptr_t)ldst;
  unsigned long long gptr = (unsigned long long)(uintptr_t)gsrc;
  asm volatile("global_load_async_to_lds_b128 %0, %1, off th:TH_LOAD_LU"
               :: "v"(lds), "v"(gptr) : "memory");
}
template <int NN>
__device__ __forceinline__ void wait_async() {
  asm volatile("s_wait_asynccnt %0" :: "i"(NN) : "memory");
}
// Constant-foldable runtime wrapper (tile loops are fully unrolled).
__device__ __forceinline__ void wait_async_n(int n) {
  switch (n) {
    case 0: wait_async<0>(); break;
    case 1: wait_async<1>(); break;
    case 2: wait_async<2>(); break;
    case 3: wait_async<3>(); break;
    case 4: wait_async<4>(); break;
    case 5: wait_async<5>(); break;
    default: wait_async<6>(); break;
  }
}

// ---- CDNA5 Tensor Data Mover: 1-D tile DMA global->LDS (TENSORcnt) ---------
typedef unsigned int u32x4 __attribute__((ext_vector_type(4)));
typedef unsigned int u32x8 __attribute__((ext_vector_type(8)));

__device__ __forceinline__ void tdm_load_1d(const float* gsrc, const float* ldst,
                                            unsigned nelem) {
  unsigned long long ga  = (unsigned long long)(uintptr_t)gsrc;
  unsigned           lds = (unsigned)(uintptr_t)ldst;
  u32x4 g0;
  g0.x = 1u;                                                  // count=1 (valid user D#)
  g0.y = lds;                                                 // lds_addr (bytes)
  g0.z = (unsigned)ga;                                        // global_addr[31:0]
  g0.w = (unsigned)((ga >> 32) & 0x01FFFFFFu) | 0x80000000u;  // addr[56:32] | type=2
  u32x8 g1;
  g1.s0 = 0x00020000u;                  // workgroup_mask=0, data_size=2 (4 bytes)
  g1.s1 = (nelem & 0xFFFFu) << 16;      // tensor_dim0[15:0]
  g1.s2 = (1u << 16) | (nelem >> 16);   // tensor_dim1=1 | tensor_dim0[31:16]
  g1.s3 = (nelem & 0xFFFFu) << 16;      // tile_dim0 | tensor_dim1[31:16]=0
  g1.s4 = 1u;                           // tile_dim2=0 | tile_dim1=1
  g1.s5 = nelem;                        // tensor_dim0_stride[31:0]
  g1.s6 = 0u;                           // stride1[15:0]<<16 | stride0[47:32]
  g1.s7 = 0u;                           // stride1[47:16]
  asm volatile("tensor_load_to_lds %0, %1" :: "s"(g0), "s"(g1) : "memory");
}
template <int NN>
__device__ __forceinline__ void wait_tensor() {
  asm volatile("s_wait_tensorcnt %0" :: "i"(NN) : "memory");
}

// ---- online-softmax (max, sumexp) helpers ----------------------------------
__device__ __forceinline__ void omerge(float& m, float& s, float m2, float s2) {
  float M = fmaxf(m, m2);
  float a = (s  == 0.f) ? 0.f : s  * __expf(m  - M);
  float b = (s2 == 0.f) ? 0.f : s2 * __expf(m2 - M);
  m = M; s = a + b;
}
__device__ __forceinline__ void oacc(float& m, float& s, float x) {
  if (x > m) { s = s * __expf(m - x) + 1.f; m = x; }
  else       { s += __expf(x - m); }
}
// Redundantly (per thread) merge the CH per-chunk partials of one row.
__device__ __forceinline__ void merge_row(const float2* __restrict__ p,
                                          float& M, float& S) {
  float m = -__builtin_inff(), s = 0.f;
  #pragma unroll 4
  for (int c = 0; c < CH; ++c) { float2 q = p[c]; omerge(m, s, q.x, q.y); }
  M = m; S = s;
}
// Workgroup (m,s) tree reduction; result valid in thread 0.
__device__ __forceinline__ void block_reduce_ms(float& m, float& s,
                                                float* sm, float* ss, int tid) {
  sm[tid] = m; ss[tid] = s; __syncthreads();
  for (int off = TPB / 2; off > 0; off >>= 1) {
    if (tid < off) { omerge(m, s, sm[tid + off], ss[tid + off]); sm[tid] = m; ss[tid] = s; }
    __syncthreads();
  }
}

// ---- Pass 0: per-chunk (max, sumexp) of logits via TDM DMA staging ---------
__global__ __launch_bounds__(TPB) void k_rowstats(const float* __restrict__ x,
                                                  float2* __restrict__ pout) {
  __shared__ float lb[2][TILE];
  __shared__ float sm[TPB], ss[TPB];
  const int row = blockIdx.x / CH, ch = blockIdx.x % CH, tid = threadIdx.x;
  const float* base = x + (size_t)row * Ncols + (size_t)ch * CHUNK;
  const bool wave0 = (tid < 32);           // one wave drives the TDM
  float m = -__builtin_inff(), s = 0.f;
  if (wave0) tdm_load_1d(base, &lb[0][0], TILE);
  #pragma unroll
  for (int t = 0; t < NT; ++t) {
    if (wave0) {
      if (t + 1 < NT) { tdm_load_1d(base + (t + 1) * TILE, &lb[(t + 1) & 1][0], TILE);
                        wait_tensor<1>(); }
      else            { wait_tensor<0>(); }
    }
    __syncthreads();                       // tile t visible to all waves
    f32x4 v = *(const f32x4*)(&lb[t & 1][tid * 4]);
    oacc(m, s, v[0]); oacc(m, s, v[1]); oacc(m, s, v[2]); oacc(m, s, v[3]);
    __syncthreads();                       // buffer free before re-DMA
  }
  block_reduce_ms(m, s, sm, ss, tid);
  if (tid == 0) pout[(size_t)row * CH + ch] = make_float2(m, s);
}

// ---- Pass 1: keys0 = logits - lse + g (stored), stats of keys0 -------------
__global__ __launch_bounds__(TPB) void k_keys0(const float* __restrict__ logits,
                                               const float* __restrict__ gno,
                                               const float2* __restrict__ pl,
                                               float* __restrict__ keys,
                                               float2* __restrict__ pout) {
  __shared__ float lbL[NBUF][TILE], lbG[NBUF][TILE];
  __shared__ float sm[TPB], ss[TPB];
  const int row = blockIdx.x / CH, ch = blockIdx.x % CH, tid = threadIdx.x;
  float Ml, Sl; merge_row(pl + (size_t)row * CH, Ml, Sl);
  const float lse = Ml + __logf(Sl);
  const size_t base = (size_t)row * Ncols + (size_t)ch * CHUNK;
  const float* bl = logits + base;
  const float* bg = gno + base;
  float* bk = keys + base;
  float m = -__builtin_inff(), s = 0.f;
  #pragma unroll
  for (int p = 0; p < DEPTH; ++p) {
    async_b128(bl + p * TILE + tid * 4, &lbL[p][tid * 4]);
    async_b128(bg + p * TILE + tid * 4, &lbG[p][tid * 4]);
  }
  #pragma unroll
  for (int t = 0; t < NT; ++t) {
    if (t + DEPTH < NT) {
      async_b128(bl + (t + DEPTH) * TILE + tid * 4, &lbL[(t + DEPTH) % NBUF][tid * 4]);
      async_b128(bg + (t + DEPTH) * TILE + tid * 4, &lbG[(t + DEPTH) % NBUF][tid * 4]);
    }
    const int inflight = (NT - 1 - t < DEPTH) ? (NT - 1 - t) : DEPTH;
    wait_async_n(2 * inflight);
    f32x4 L = *(const f32x4*)(&lbL[t % NBUF][tid * 4]);
    f32x4 G = *(const f32x4*)(&lbG[t % NBUF][tid * 4]);
    f32x4 kv;
    #pragma unroll
    for (int j = 0; j < 4; ++j) {
      float k = L[j] - lse + G[j];
      kv[j] = k; oacc(m, s, k);
    }
    *(f32x4*)(bk + t * TILE + tid * 4) = kv;
  }
  block_reduce_ms(m, s, sm, ss, tid);
  if (tid == 0) pout[(size_t)row * CH + ch] = make_float2(m, s);
}

// ---- Passes 2..16: in-place keys update + stats of new keys ----------------
__global__ __launch_bounds__(TPB) void k_iter(float* __restrict__ keys,
                                              const float2* __restrict__ pin,
                                              float2* __restrict__ pout) {
  __shared__ float lb[NBUF][TILE];
  __shared__ float sm[TPB], ss[TPB];
  const int row = blockIdx.x / CH, ch = blockIdx.x % CH, tid = threadIdx.x;
  float M, Z; merge_row(pin + (size_t)row * CH, M, Z);
  const float invZ = 1.f / Z;
  float* bk = keys + (size_t)row * Ncols + (size_t)ch * CHUNK;
  float m = -__builtin_inff(), s = 0.f;
  #pragma unroll
  for (int p = 0; p < DEPTH; ++p)
    async_b128(bk + p * TILE + tid * 4, &lb[p][tid * 4]);
  #pragma unroll
  for (int t = 0; t < NT; ++t) {
    if (t + DEPTH < NT)
      async_b128(bk + (t + DEPTH) * TILE + tid * 4, &lb[(t + DEPTH) % NBUF][tid * 4]);
    const int inflight = (NT - 1 - t < DEPTH) ? (NT - 1 - t) : DEPTH;
    wait_async_n(inflight);
    f32x4 k0 = *(const f32x4*)(&lb[t % NBUF][tid * 4]);
    f32x4 kv;
    #pragma unroll
    for (int j = 0; j < 4; ++j) {
      float x = k0[j];
      float p = __expf(x - M) * invZ;                // softmax(keys) element
      float k = x + __logf(fmaxf(1.f - p, EPS));
      kv[j] = k; oacc(m, s, k);
    }
    *(f32x4*)(bk + t * TILE + tid * 4) = kv;
  }
  block_reduce_ms(m, s, sm, ss, tid);
  if (tid == 0) pout[(size_t)row * CH + ch] = make_float2(m, s);
}

// ---- Final sweep: replay all K iterations per element, write khot,
// ---- zero pert, extract per-chunk top-16 candidates ------------------------
__global__ __launch_bounds__(TPB) void k_final(const float* __restrict__ logits,
                                               const float* __restrict__ gno,
                                               const float2* __restrict__ part, // set0=lse, sets1..16=keys_t
                                               float* __restrict__ out_khot,
                                               float* __restrict__ out_pert,
                                               float* __restrict__ cval,
                                               int* __restrict__ cidx) {
  __shared__ float lbL[2][TILE], lbG[2][TILE];
  __shared__ float kh[CHUNK];                 // 32 KB chunk khot
  __shared__ float rv[TPB];
  __shared__ int   ri[TPB];
  const int row = blockIdx.x / CH, ch = blockIdx.x % CH, tid = threadIdx.x;

  float Ml, Sl; merge_row(part + (size_t)row * CH, Ml, Sl);
  const float lse = Ml + __logf(Sl);
  float mt[Ksel], iz[Ksel];
  #pragma unroll
  for (int t = 0; t < Ksel; ++t) {
    float M, Z; merge_row(part + (size_t)(t + 1) * Brows * CH + (size_t)row * CH, M, Z);
    mt[t] = M; iz[t] = 1.f / Z;
  }

  const size_t base = (size_t)row * Ncols + (size_t)ch * CHUNK;
  const float* bl = logits + base;
  const float* bg = gno + base;
  async_b128_lu(bl + tid * 4, &lbL[0][tid * 4]);
  async_b128_lu(bg + tid * 4, &lbG[0][tid * 4]);
  #pragma unroll
  for (int t = 0; t < NT; ++t) {
    if (t + 1 < NT) {
      async_b128_lu(bl + (t + 1) * TILE + tid * 4, &lbL[(t + 1) & 1][tid * 4]);
      async_b128_lu(bg + (t + 1) * TILE + tid * 4, &lbG[(t + 1) & 1][tid * 4]);
      wait_async<2>();
    } else wait_async<0>();
    f32x4 L = *(const f32x4*)(&lbL[t & 1][tid * 4]);
    f32x4 G = *(const f32x4*)(&lbG[t & 1][tid * 4]);
    f32x4 kv;
    #pragma unroll
    for (int j = 0; j < 4; ++j) {
      float k   = L[j] - lse + G[j];
      float acc = 0.f;
      #pragma unroll
      for (int tt = 0; tt < Ksel; ++tt) {
        float p = __expf(k - mt[tt]) * iz[tt];
        acc += p;
        k += __logf(fmaxf(1.f - p, EPS));
      }
      kv[j] = acc;
    }
    const int cb = t * TILE + tid * 4;
    kh[cb + 0] = kv[0]; kh[cb + 1] = kv[1]; kh[cb + 2] = kv[2]; kh[cb + 3] = kv[3];
    // Write-once outputs: non-temporal, keep the 192MB L2 for logits+g+keys.
    __builtin_nontemporal_store(kv, (f32x4*)(out_khot + base + cb));
    f32x4 z = {0.f, 0.f, 0.f, 0.f};
    __builtin_nontemporal_store(z, (f32x4*)(out_pert + base + cb));
  }
  __syncthreads();

  // 16 rounds of argmax-with-exclusion over the 8192-element chunk in LDS.
  for (int r = 0; r < Ksel; ++r) {
    float bv = -1.f; int bi = CHUNK;
    for (int i = tid; i < CHUNK; i += TPB) {
      float v = kh[i];
      if (v > bv || (v == bv && i < bi)) { bv = v; bi = i; }
    }
    rv[tid] = bv; ri[tid] = bi; __syncthreads();
    for (int off = TPB / 2; off > 0; off >>= 1) {
      if (tid < off) {
        float v2 = rv[tid + off]; int i2 = ri[tid + off];
        if (v2 > rv[tid] || (v2 == rv[tid] && i2 < ri[tid])) { rv[tid] = v2; ri[tid] = i2; }
      }
      __syncthreads();
    }
    if (tid == 0) {
      const int slot = (row * CH + ch) * Ksel + r;
      cval[slot] = rv[0];
      cidx[slot] = ch * CHUNK + ri[0];
      kh[ri[0]] = -__builtin_inff();
    }
    __syncthreads();
  }
}

// ---- Per-row global top-16 from 32x16 candidates; scatter the ones ---------
__global__ __launch_bounds__(TPB) void k_scatter(const float* __restrict__ cval,
                                                 const int* __restrict__ cidx,
                                                 float* __restrict__ out_pert) {
  constexpr int NC = CH * Ksel;  // 512 candidates per row
  __shared__ float cv[NC];
  __shared__ int   ci[NC];
  __shared__ float rv[TPB];
  __shared__ int   rj[TPB];
  const int row = blockIdx.x, tid = threadIdx.x;
  for (int i = tid; i < NC; i += TPB) {
    cv[i] = cval[(size_t)row * NC + i];
    ci[i] = cidx[(size_t)row * NC + i];
  }
  __syncthreads();
  for (int r = 0; r < Ksel; ++r) {
    float bv = -__builtin_inff(); int bj = -1; int bc = 0x7fffffff;
    for (int i = tid; i < NC; i += TPB) {
      float v = cv[i]; int c = ci[i];
      if (v > bv || (v == bv && c < bc)) { bv = v; bj = i; bc = c; }
    }
    rv[tid] = bv; rj[tid] = bj; __syncthreads();
    for (int off = TPB / 2; off > 0; off >>= 1) {
      if (tid < off) {
        float v2 = rv[tid + off]; int j2 = rj[tid + off];
        int c1 = (rj[tid] >= 0) ? ci[rj[tid]] : 0x7fffffff;
        int c2 = (j2 >= 0) ? ci[j2] : 0x7fffffff;
        if (v2 > rv[tid] || (v2 == rv[tid] && c2 < c1)) { rv[tid] = v2; rj[tid] = j2; }
      }
      __syncthreads();
    }
    if (tid == 0) {
      int j = rj[0];
      if (j >= 0) { out_pert[(size_t)row * Ncols + ci[j]] = 1.0f; cv[j] = -__builtin_inff(); }
    }
    __syncthreads();
  }
}

extern "C" void kernel_launch(void* const* d_in, const int* in_sizes, int n_in,
                              void* d_out, int out_size, void* d_ws, size_t ws_size,
                              hipStream_t stream) {
  (void)in_sizes; (void)n_in; (void)out_size; (void)ws_size;
  const float* logits = (const float*)d_in[0];
  const float* gno    = (const float*)d_in[1];
  float* out_pert = (float*)d_out;                       // [B, N] hard k-hot
  float* out_khot = out_pert + (size_t)Brows * Ncols;    // [B, N] soft khot

  // Workspace: keys (64 MB) | 17 partial sets | candidate vals | candidate idx
  char*   ws    = (char*)d_ws;
  float*  keys  = (float*)ws;
  float2* part  = (float2*)(ws + sizeof(float) * (size_t)Brows * Ncols);
  float*  cval  = (float*)(part + (size_t)(Ksel + 1) * Brows * CH);
  int*    cidx  = (int*)(cval + (size_t)Brows * CH * Ksel);

  const dim3 grid(Brows * CH), blk(TPB);
  k_rowstats<<<grid, blk, 0, stream>>>(logits, part);                       // set 0
  k_keys0<<<grid, blk, 0, stream>>>(logits, gno, part, keys,
                                    part + (size_t)Brows * CH);             // set 1
  for (int t = 1; t < Ksel; ++t)
    k_iter<<<grid, blk, 0, stream>>>(keys,
                                     part + (size_t)t * Brows * CH,
                                     part + (size_t)(t + 1) * Brows * CH);
  k_final<<<grid, blk, 0, stream>>>(logits, gno, part, out_khot, out_pert, cval, cidx);
  k_scatter<<<dim3(Brows), blk, 0, stream>>>(cval, cidx, out_pert);
}